// QuantumGenerator_66743791780059
// MI455X (gfx1250) — compile-verified
//
#include <hip/hip_runtime.h>

typedef __attribute__((ext_vector_type(2))) float v2f;
typedef __attribute__((ext_vector_type(8))) float v8f;

#define NQ      5
#define DIM     32            // 2^NQ state dimension
#define NLAYERS 6
#define NGEN    4
#define OUTCOLS 64            // 4 generators * 16 patch columns

// ---------------------------------------------------------------------------
// Kernel 1: fold the 6 batch-independent weight layers into one 32x32 matrix
// per generator.  wave g simulates generator g; lane j carries column j of the
// evolving operator (applied to identity).  Fully unrolled so st[] lives in
// VGPRs.  Output: U[g*1024 + row*32 + col]  (row = output amplitude index).
// ---------------------------------------------------------------------------
__global__ void build_unitaries(const float* __restrict__ qp,
                                float* __restrict__ U) {
    const int g    = threadIdx.x >> 5;   // generator = wave id (0..3)
    const int lane = threadIdx.x & 31;   // column j of U_g

    float st[DIM];
    #pragma unroll
    for (int i = 0; i < DIM; ++i) st[i] = (i == lane) ? 1.0f : 0.0f;

    #pragma unroll
    for (int l = 0; l < NLAYERS; ++l) {
        #pragma unroll
        for (int w = 0; w < NQ; ++w) {
            const float th = 0.5f * qp[g * (NLAYERS * NQ) + l * NQ + w];
            const float c = cosf(th);
            const float s = sinf(th);
            const int bit = 1 << (NQ - 1 - w);   // wire 0 = MSB (axis 0)
            #pragma unroll
            for (int i = 0; i < DIM; ++i) {
                if (!(i & bit)) {
                    const float a0 = st[i];
                    const float a1 = st[i | bit];
                    st[i]       = c * a0 - s * a1;
                    st[i | bit] = s * a0 + c * a1;
                }
            }
        }
        // CZ chain: sign = (-1)^(# adjacent bit pairs set) =
        // parity of popcount(i & (i>>1) & 0xF).
        #pragma unroll
        for (int i = 0; i < DIM; ++i) {
            if (__popc(i & (i >> 1) & 0xF) & 1) st[i] = -st[i];
        }
    }

    #pragma unroll
    for (int i = 0; i < DIM; ++i)
        U[g * (DIM * DIM) + i * DIM + lane] = st[i];
}

// s0[k] for a product state: prod over wires of cos/sin(x_w/2)
__device__ __forceinline__ float s0val(int k, const float* c, const float* s) {
    float p = 1.0f;
    #pragma unroll
    for (int w = 0; w < NQ; ++w)
        p *= ((k >> (NQ - 1 - w)) & 1) ? s[w] : c[w];
    return p;
}

// Single-instruction xor-exchange: DS_SWIZZLE_B32, group-of-32 mode,
// and_mask=0x1f, or_mask=0, xor_mask=M.  Masks <= 8 stay inside each
// 16-lane half, matching the f32 D-tile layout (N = lane%16).
template <int M>
__device__ __forceinline__ float swz_xor(float v) {
    return __int_as_float(
        __builtin_amdgcn_ds_swizzle(__float_as_int(v), (M << 10) | 0x1f));
}

// ---------------------------------------------------------------------------
// Kernel 2: per wave, 16 batch rows.  Out(16x16) = S0(16x32) x U_g^T(32x16)
// via 8 chained V_WMMA_F32_16X16X4_F32, then p^2 and row-max normalization.
// (probs[:, :16]/sum then /max algebraically equals p_j / max_{j<16} p_j,
//  so neither the sum nor columns 16..31 are ever needed.)
// ---------------------------------------------------------------------------
__global__ void __launch_bounds__(256)
qgen_main(const float* __restrict__ x, const float* __restrict__ U,
          float* __restrict__ out, int B) {
    __shared__ float Ul[NGEN * DIM * DIM];           // 16 KB of 320 KB LDS
    for (int i = threadIdx.x; i < NGEN * DIM * DIM; i += blockDim.x)
        Ul[i] = U[i];
    __syncthreads();

    const int lane = threadIdx.x & 31;
    const int wave = threadIdx.x >> 5;
    const int hi   = lane >> 4;        // lane half selects K pair / M half
    const int lm   = lane & 15;        // M (for A) and N (for B/D) index
    const long tile    = (long)blockIdx.x * (blockDim.x >> 5) + wave;
    const long rowBase = tile * 16;
    if (rowBase >= B) return;          // wave-uniform exit: EXEC stays all-1s
    const bool full = (rowBase + 16 <= (long)B);   // wave-uniform

    // ---- A fragments: s0 for batch row rowBase+lm ----
    long r = rowBase + lm;
    if (r >= B) r = B - 1;             // clamp reads, keep EXEC full for WMMA
    float c[NQ], s[NQ];
    #pragma unroll
    for (int w = 0; w < NQ; ++w) {
        const float th = 0.5f * x[r * NQ + w];
        c[w] = cosf(th);
        s[w] = sinf(th);
    }
    // ISA A layout (16x4 f32): VGPR0 -> K = 4*kk + 2*hi, VGPR1 -> +1
    float a0[8], a1[8];
    #pragma unroll
    for (int kk = 0; kk < 8; ++kk) {
        const int k0 = 4 * kk + 2 * hi;
        a0[kk] = s0val(k0,     c, s);
        a1[kk] = s0val(k0 + 1, c, s);
    }

    // Lane-fixed output base: row = rowBase + 8*hi + v, col = g*16 + lm.
    // All g/v offsets are compile-time -> global_store_b32 immediate offsets.
    float* const op = out + (size_t)(rowBase + 8 * hi) * OUTCOLS + lm;

    #pragma unroll
    for (int g = 0; g < NGEN; ++g) {
        const float* Ug = &Ul[g * DIM * DIM];
        v8f acc = {};
        #pragma unroll
        for (int kk = 0; kk < 8; ++kk) {
            const int k0 = 4 * kk + 2 * hi;
            v2f A = {a0[kk], a1[kk]};
            // B[k][n] = U^T[k][n] = U[n][k]; N = lm, K split mirrors A
            v2f Bv = {Ug[lm * DIM + k0], Ug[lm * DIM + k0 + 1]};
            acc = __builtin_amdgcn_wmma_f32_16x16x4_f32(
                false, A, false, Bv, (short)0, acc, false, false);
        }

        // D layout: VGPR v -> row m = v + 8*hi, col n = lm.
        // probs = amp^2; rowmax via 4 swizzle stages, all 8 rows in flight
        // per stage so the LDS-return latency of ds_swizzle is overlapped.
        float p[8], m[8], t[8];
        #pragma unroll
        for (int v = 0; v < 8; ++v) { p[v] = acc[v] * acc[v]; m[v] = p[v]; }
        #pragma unroll
        for (int v = 0; v < 8; ++v) t[v] = swz_xor<1>(m[v]);
        #pragma unroll
        for (int v = 0; v < 8; ++v) m[v] = fmaxf(m[v], t[v]);
        #pragma unroll
        for (int v = 0; v < 8; ++v) t[v] = swz_xor<2>(m[v]);
        #pragma unroll
        for (int v = 0; v < 8; ++v) m[v] = fmaxf(m[v], t[v]);
        #pragma unroll
        for (int v = 0; v < 8; ++v) t[v] = swz_xor<4>(m[v]);
        #pragma unroll
        for (int v = 0; v < 8; ++v) m[v] = fmaxf(m[v], t[v]);
        #pragma unroll
        for (int v = 0; v < 8; ++v) t[v] = swz_xor<8>(m[v]);
        #pragma unroll
        for (int v = 0; v < 8; ++v) m[v] = fmaxf(m[v], t[v]);

        if (full) {
            #pragma unroll
            for (int v = 0; v < 8; ++v)
                op[g * 16 + v * OUTCOLS] = p[v] / m[v];
        } else {
            #pragma unroll
            for (int v = 0; v < 8; ++v) {
                const long row = rowBase + v + 8 * hi;
                if (row < B)
                    out[row * OUTCOLS + g * 16 + lm] = p[v] / m[v];
            }
        }
    }
}

extern "C" void kernel_launch(void* const* d_in, const int* in_sizes, int n_in,
                              void* d_out, int out_size, void* d_ws, size_t ws_size,
                              hipStream_t stream) {
    const float* x  = (const float*)d_in[0];   // (B, 5)
    const float* qp = (const float*)d_in[1];   // (4, 30)
    float* out = (float*)d_out;                // (B, 64)
    float* U   = (float*)d_ws;                 // 4 * 32 * 32 floats = 16 KB
    const int B = in_sizes[0] / NQ;

    hipLaunchKernelGGL(build_unitaries, dim3(1), dim3(NGEN * 32), 0, stream, qp, U);

    const int tiles  = (B + 15) / 16;
    const int blocks = (tiles + 7) / 8;        // 8 waves (tiles) per block
    hipLaunchKernelGGL(qgen_main, dim3(blocks), dim3(256), 0, stream, x, U, out, B);
}